// TemporalGCN_42734924595843
// MI455X (gfx1250) — compile-verified
//
#include <hip/hip_runtime.h>
#include <math.h>

typedef __attribute__((ext_vector_type(2))) float v2f;
typedef __attribute__((ext_vector_type(8))) float v8f;

#define LSEQ 512
#define DIM  128
#define DHID 256

// ---------------- Kernel 1: per-row inverse L2 norms ----------------
__global__ __launch_bounds__(128) void gcn_norm_kernel(const float* __restrict__ x,
                                                       float* __restrict__ invn,
                                                       int nrows) {
    int wave = blockIdx.x * (blockDim.x >> 5) + (threadIdx.x >> 5);
    int lane = threadIdx.x & 31;
    if (wave >= nrows) return;
    const float* row = x + (size_t)wave * DIM;
    float ss = 0.f;
    #pragma unroll
    for (int i = 0; i < DIM / 32; ++i) {
        float v = row[lane + i * 32];
        ss += v * v;
    }
    #pragma unroll
    for (int off = 16; off > 0; off >>= 1)
        ss += __shfl_xor(ss, off, 32);
    if (lane == 0)
        invn[wave] = 1.0f / fmaxf(sqrtf(ss), 1e-12f);
}

// ---------------- Kernel 2: fused sim->topk->agg->MLP ----------------
__global__ __launch_bounds__(128) void gcn_main_kernel(
    const float* __restrict__ x, const float* __restrict__ invn,
    const float* __restrict__ W,  const float* __restrict__ bb,
    const float* __restrict__ W1, const float* __restrict__ b1,
    const float* __restrict__ W2, const float* __restrict__ b2,
    const float* __restrict__ gamma, const float* __restrict__ beta,
    float* __restrict__ out)
{
    const int bj = blockIdx.x >> 5;   // 192 batches
    const int rb = blockIdx.x & 31;   // 32 row-blocks of 16
    const int r0 = rb * 16;
    const float* xb  = x    + (size_t)bj * LSEQ * DIM;
    const float* inv = invn + (size_t)bj * LSEQ;

    __shared__ float sA[16 * 129];    // normalized A rows (padded stride)
    __shared__ float sAgg[16 * 129];  // aggregated neighbors
    __shared__ float sBig[16 * 520];  // sim[16][520]; reused: h(16x130) + act(16x258)
    __shared__ int   sIdx[16 * 8];

    const int tid  = threadIdx.x;
    const int lane = tid & 31;
    const int wave = tid >> 5;
    const int m    = lane & 15;
    const int koff = (lane >> 4) << 1;   // 0 or 2
    const int Mb   = (lane >> 4) << 3;   // 0 or 8 (C/D row base)

    // ---- stage 1: stage normalized row-tile for the A operand ----
    for (int i = tid; i < 16 * DIM; i += 128) {
        int r = i >> 7, k = i & 127;
        sA[r * 129 + k] = xb[(size_t)(r0 + r) * DIM + k] * inv[r0 + r];
    }
    __syncthreads();

    // ---- stage 2: sim tile row-block vs all 512 columns (fp32 WMMA) ----
    {
        const int arow = m * 129;
        for (int ct = wave; ct < 32; ct += 4) {
            const int c0 = ct * 16;
            const float* xc = xb + (size_t)(c0 + m) * DIM;
            const float  ic = inv[c0 + m];
            v8f acc = {};
            for (int kb = 0; kb < DIM; kb += 4) {
                v2f a, bf;
                a[0]  = sA[arow + kb + koff];
                a[1]  = sA[arow + kb + koff + 1];
                bf[0] = xc[kb + koff] * ic;
                bf[1] = xc[kb + koff + 1] * ic;
                acc = __builtin_amdgcn_wmma_f32_16x16x4_f32(
                        false, a, false, bf, (short)0, acc, false, false);
            }
            #pragma unroll
            for (int v = 0; v < 8; ++v)
                sBig[(Mb + v) * 520 + c0 + m] = acc[v];
        }
    }
    __syncthreads();

    // ---- stage 3: stable top-8 per row (ties -> lower index, like top_k) ----
    if (tid < 16) {
        float vk[8]; int ik[8];
        #pragma unroll
        for (int i = 0; i < 8; ++i) { vk[i] = -__builtin_inff(); ik[i] = 0; }
        const float* srow = &sBig[tid * 520];
        for (int j = 0; j < LSEQ; ++j) {
            float s = srow[j];
            if (s > vk[7]) {
                int p = 7;
                while (p > 0 && s > vk[p - 1]) {
                    vk[p] = vk[p - 1]; ik[p] = ik[p - 1]; --p;
                }
                vk[p] = s; ik[p] = j;
            }
        }
        #pragma unroll
        for (int i = 0; i < 8; ++i) sIdx[tid * 8 + i] = ik[i];
    }
    __syncthreads();

    // ---- stage 4: x_agg = mean of 8 gathered raw rows ----
    for (int r = 0; r < 16; ++r) {
        float s = 0.f;
        #pragma unroll
        for (int i = 0; i < 8; ++i)
            s += xb[(size_t)sIdx[r * 8 + i] * DIM + tid];
        sAgg[r * 129 + tid] = s * 0.125f;
    }
    __syncthreads();

    float* hbuf = sBig;              // 16 x 130
    float* abuf = sBig + 16 * 130;   // 16 x 258

    // ---- stage 5: h = agg @ W + b  (WMMA, K=128) ----
    {
        const int arow = m * 129;
        for (int ct = wave; ct < 8; ct += 4) {
            const int n0 = ct * 16;
            v8f acc = {};
            for (int kb = 0; kb < DIM; kb += 4) {
                v2f a, bf;
                a[0] = sAgg[arow + kb + koff];
                a[1] = sAgg[arow + kb + koff + 1];
                const float* wp = W + (size_t)(kb + koff) * DIM + n0 + m;
                bf[0] = wp[0];
                bf[1] = wp[DIM];
                acc = __builtin_amdgcn_wmma_f32_16x16x4_f32(
                        false, a, false, bf, (short)0, acc, false, false);
            }
            const float bias = bb[n0 + m];
            #pragma unroll
            for (int v = 0; v < 8; ++v)
                hbuf[(Mb + v) * 130 + n0 + m] = acc[v] + bias;
        }
    }
    __syncthreads();

    // ---- stage 6: LayerNorm per row ----
    if (tid < 16) {
        float* hr = hbuf + tid * 130;
        float mu = 0.f, s2 = 0.f;
        for (int n = 0; n < DIM; ++n) { float v = hr[n]; mu += v; s2 += v * v; }
        mu *= (1.0f / DIM);
        s2 = s2 * (1.0f / DIM) - mu * mu;
        float rstd = rsqrtf(s2 + 1e-5f);
        for (int n = 0; n < DIM; ++n)
            hr[n] = (hr[n] - mu) * rstd * gamma[n] + beta[n];
    }
    __syncthreads();

    // ---- stage 7: act = relu(hn @ W1 + b1)  (K=128, 256 outputs) ----
    {
        const int arow = m * 130;
        for (int ct = wave; ct < 16; ct += 4) {
            const int n0 = ct * 16;
            v8f acc = {};
            for (int kb = 0; kb < DIM; kb += 4) {
                v2f a, bf;
                a[0] = hbuf[arow + kb + koff];
                a[1] = hbuf[arow + kb + koff + 1];
                const float* wp = W1 + (size_t)(kb + koff) * DHID + n0 + m;
                bf[0] = wp[0];
                bf[1] = wp[DHID];
                acc = __builtin_amdgcn_wmma_f32_16x16x4_f32(
                        false, a, false, bf, (short)0, acc, false, false);
            }
            const float bias = b1[n0 + m];
            #pragma unroll
            for (int v = 0; v < 8; ++v)
                abuf[(Mb + v) * 258 + n0 + m] = fmaxf(acc[v] + bias, 0.f);
        }
    }
    __syncthreads();

    // ---- stage 8: out = act @ W2 + b2  (K=256) -> global ----
    {
        const int arow = m * 258;
        for (int ct = wave; ct < 8; ct += 4) {
            const int n0 = ct * 16;
            v8f acc = {};
            for (int kb = 0; kb < DHID; kb += 4) {
                v2f a, bf;
                a[0] = abuf[arow + kb + koff];
                a[1] = abuf[arow + kb + koff + 1];
                const float* wp = W2 + (size_t)(kb + koff) * DIM + n0 + m;
                bf[0] = wp[0];
                bf[1] = wp[DIM];
                acc = __builtin_amdgcn_wmma_f32_16x16x4_f32(
                        false, a, false, bf, (short)0, acc, false, false);
            }
            const float bias = b2[n0 + m];
            float* op = out + ((size_t)bj * LSEQ + r0 + Mb) * DIM + n0 + m;
            #pragma unroll
            for (int v = 0; v < 8; ++v)
                op[(size_t)v * DIM] = acc[v] + bias;
        }
    }
}

extern "C" void kernel_launch(void* const* d_in, const int* in_sizes, int n_in,
                              void* d_out, int out_size, void* d_ws, size_t ws_size,
                              hipStream_t stream) {
    (void)in_sizes; (void)n_in; (void)out_size; (void)ws_size;
    const float* x     = (const float*)d_in[0];
    const float* W     = (const float*)d_in[1];
    const float* b     = (const float*)d_in[2];
    const float* W1    = (const float*)d_in[3];
    const float* b1    = (const float*)d_in[4];
    const float* W2    = (const float*)d_in[5];
    const float* b2    = (const float*)d_in[6];
    const float* gamma = (const float*)d_in[7];
    const float* beta  = (const float*)d_in[8];
    float* out  = (float*)d_out;
    float* invn = (float*)d_ws;   // 192*512 floats = 384 KB

    const int nrows = 192 * LSEQ;
    gcn_norm_kernel<<<nrows / 4, 128, 0, stream>>>(x, invn, nrows);
    gcn_main_kernel<<<192 * 32, 128, 0, stream>>>(x, invn, W, b, W1, b1, W2, b2,
                                                  gamma, beta, out);
}